// Histogram_Matching_60361470378361
// MI455X (gfx1250) — compile-verified
//
#include <hip/hip_runtime.h>

#define NBINS 256

typedef __attribute__((ext_vector_type(2))) float v2f;
typedef __attribute__((ext_vector_type(8))) float v8f;

// ---------------------------------------------------------------------------
// Kernel 0: zero the histogram workspace (graph replays require re-init).
// ---------------------------------------------------------------------------
__global__ void hm_zero_kernel(unsigned int* __restrict__ ws, int n) {
    int i = blockIdx.x * blockDim.x + threadIdx.x;
    if (i < n) ws[i] = 0u;
}

// ---------------------------------------------------------------------------
// Kernel 1: per-channel 256-bin histograms of dst and ref.
// LDS-privatized histogram per block, one global atomic per bin per block.
// idx = clip(floor((img*255) * 256/255), 0, 255)  -- torch.histc semantics.
// ---------------------------------------------------------------------------
__global__ __launch_bounds__(256) void hm_hist_kernel(
        const float* __restrict__ dst, const float* __restrict__ ref,
        unsigned int* __restrict__ histD, unsigned int* __restrict__ histR,
        int vec4PerThread) {
    __shared__ unsigned int lh[NBINS];
    const int t = threadIdx.x;
    lh[t] = 0u;
    __syncthreads();

    const float* img = (blockIdx.z == 0) ? dst : ref;
    unsigned int* hist = (blockIdx.z == 0) ? histD : histR;
    const int ch = blockIdx.y;

    const size_t pxPerBlock = (size_t)vec4PerThread * 256u * 4u;
    const size_t HW = (size_t)gridDim.x * pxPerBlock;
    const float4* p = (const float4*)(img + (size_t)ch * HW) +
                      (size_t)blockIdx.x * ((size_t)vec4PerThread * 256u);

    for (int i = 0; i < vec4PerThread; ++i) {
        float4 v = p[t + i * 256];
        // speculative streaming prefetch (global_prefetch_b8)
        __builtin_prefetch((const void*)&p[t + (i + 3) * 256], 0, 1);
        float xs[4] = {v.x, v.y, v.z, v.w};
#pragma unroll
        for (int k = 0; k < 4; ++k) {
            float x = xs[k] * 255.0f;
            int idx = (int)(x * (256.0f / 255.0f));
            idx = idx < 0 ? 0 : (idx > 255 ? 255 : idx);
            atomicAdd(&lh[idx], 1u);
        }
    }
    __syncthreads();
    atomicAdd(&hist[ch * NBINS + t], lh[t]);
}

// ---------------------------------------------------------------------------
// WMMA cumsum: 256-entry CDF as one 16x16 f32 tile, K=256 contraction via
// 128 chained V_WMMA_F32_16X16X4_F32 ops. D[i][j] = cumsum[16*i + j]:
//   D = A1 x B1 + A2 x B2
//   A1[m][k] = (k < 16*m)           B1[k][n] = h[k]          (row-block offsets)
//   A2[m][k] = ((k>>4)==m) ? h[k]:0 B2[k][n] = ((k&15) <= n) (within-row prefix)
// Register layouts per CDNA5 ISA 7.12.2 (wave32):
//   A 16x4: lane m / m+16 ; VGPR0 = K{k0, k0+2}, VGPR1 = K{k0+1, k0+3}
//   B 4x16: lane = N ; same K interleave. C/D: lane = N, VGPR p -> M = p + 8*hi.
// ---------------------------------------------------------------------------
__device__ __forceinline__ v8f hm_cdf_wmma(const float* __restrict__ h,
                                           int m /*laneLo*/, int hi) {
    v8f c = {};
    // Part 1: offsets of preceding full rows.
    for (int kb = 0; kb < 64; ++kb) {
        int k0 = 4 * kb + 2 * hi;
        v2f a, b;
        a.x = (k0     < 16 * m) ? 1.0f : 0.0f;
        a.y = (k0 + 1 < 16 * m) ? 1.0f : 0.0f;
        b.x = h[k0];          // independent of N: uniform LDS read
        b.y = h[k0 + 1];
        c = __builtin_amdgcn_wmma_f32_16x16x4_f32(false, a, false, b,
                                                  (short)0, c, false, false);
    }
    // Part 2: within-row prefix (triu).
    for (int kb = 0; kb < 64; ++kb) {
        int k0 = 4 * kb + 2 * hi;
        v2f a, b;
        a.x = ((k0 >> 4) == m)       ? h[k0]     : 0.0f;
        a.y = (((k0 + 1) >> 4) == m) ? h[k0 + 1] : 0.0f;
        b.x = ((k0 & 15)       <= m) ? 1.0f : 0.0f;   // N == laneLo == m
        b.y = (((k0 + 1) & 15) <= m) ? 1.0f : 0.0f;
        c = __builtin_amdgcn_wmma_f32_16x16x4_f32(false, a, false, b,
                                                  (short)0, c, false, false);
    }
    return c;
}

// ---------------------------------------------------------------------------
// Kernel 2: per channel (one wave32 block): normalize hists, WMMA cumsum ->
// CDFs, then tables[d] = clip(#{r: cdfD[d] >= cdfR[r]} - 1, 0, 255) / 255.
// ---------------------------------------------------------------------------
__global__ __launch_bounds__(32) void hm_tables_kernel(
        const unsigned int* __restrict__ histD,
        const unsigned int* __restrict__ histR,
        float* __restrict__ tables, float invHW) {
    __shared__ float shD[NBINS];
    __shared__ float shR[NBINS];
    __shared__ float cdR[NBINS];

    const int lane   = threadIdx.x;
    const int laneLo = lane & 15;
    const int hi     = lane >> 4;
    const int ch     = blockIdx.x;

#pragma unroll
    for (int i = 0; i < 8; ++i) {
        int d = lane * 8 + i;
        shD[d] = (float)histD[ch * NBINS + d] * invHW;
        shR[d] = (float)histR[ch * NBINS + d] * invHW;
    }
    __syncthreads();

    v8f cD = hm_cdf_wmma(shD, laneLo, hi);
    v8f cR = hm_cdf_wmma(shR, laneLo, hi);

    // Scatter ref CDF from D-layout to LDS in linear order.
#pragma unroll
    for (int p = 0; p < 8; ++p) {
        int d = 16 * (p + 8 * hi) + laneLo;
        cdR[d] = cR[p];
    }
    __syncthreads();

    float cnt[8];
#pragma unroll
    for (int p = 0; p < 8; ++p) cnt[p] = 0.0f;

    for (int r = 0; r < NBINS; ++r) {
        float rv = cdR[r];
#pragma unroll
        for (int p = 0; p < 8; ++p)
            cnt[p] += (cD[p] >= rv) ? 1.0f : 0.0f;
    }

#pragma unroll
    for (int p = 0; p < 8; ++p) {
        float tv = cnt[p] - 1.0f;
        tv = fminf(fmaxf(tv, 0.0f), 255.0f);
        int d = 16 * (p + 8 * hi) + laneLo;
        tables[ch * NBINS + d] = tv / 255.0f;   // pre-divide for apply pass
    }
}

// ---------------------------------------------------------------------------
// Kernel 3: apply LUT. Preserves the reference bug: LUT row for (b,c) is b*c.
// ---------------------------------------------------------------------------
__global__ __launch_bounds__(256) void hm_apply_kernel(
        const float* __restrict__ dst, const float* __restrict__ tables,
        float* __restrict__ out) {
    __shared__ float lut[NBINS];
    const int ch = blockIdx.y;          // 0..47
    const int b = ch / 3, c = ch % 3;
    const int lutRow = b * c;           // faithful bug
    lut[threadIdx.x] = tables[lutRow * NBINS + threadIdx.x];
    __syncthreads();

    const size_t HW = (size_t)gridDim.x * 2048u;
    const float4* in = (const float4*)(dst + (size_t)ch * HW) +
                       (size_t)blockIdx.x * 512u;
    float4* o = (float4*)(out + (size_t)ch * HW) + (size_t)blockIdx.x * 512u;

#pragma unroll
    for (int i = 0; i < 2; ++i) {
        float4 v = in[threadIdx.x + i * 256];
        float4 r;
        int ix = (int)(v.x * 255.0f); ix = ix < 0 ? 0 : (ix > 255 ? 255 : ix);
        int iy = (int)(v.y * 255.0f); iy = iy < 0 ? 0 : (iy > 255 ? 255 : iy);
        int iz = (int)(v.z * 255.0f); iz = iz < 0 ? 0 : (iz > 255 ? 255 : iz);
        int iw = (int)(v.w * 255.0f); iw = iw < 0 ? 0 : (iw > 255 ? 255 : iw);
        r.x = lut[ix]; r.y = lut[iy]; r.z = lut[iz]; r.w = lut[iw];
        o[threadIdx.x + i * 256] = r;
    }
}

// ---------------------------------------------------------------------------
extern "C" void kernel_launch(void* const* d_in, const int* in_sizes, int n_in,
                              void* d_out, int out_size, void* d_ws, size_t ws_size,
                              hipStream_t stream) {
    const float* dst = (const float*)d_in[0];
    const float* ref = (const float*)d_in[1];
    float* out = (float*)d_out;

    const int nCh = 48;                         // 16 * 3
    const int HW = in_sizes[0] / nCh;           // 1048576

    unsigned int* histD = (unsigned int*)d_ws;
    unsigned int* histR = histD + nCh * NBINS;
    float* tables = (float*)(histR + nCh * NBINS);

    // 0) zero histograms
    {
        int n = 2 * nCh * NBINS;
        hm_zero_kernel<<<(n + 255) / 256, 256, 0, stream>>>(histD, n);
    }
    // 1) histograms of dst and ref (blockIdx.z selects image)
    {
        const int vec4PerThread = 32;                       // 32768 px / block
        const int blocksPerCh = HW / (vec4PerThread * 1024);
        hm_hist_kernel<<<dim3(blocksPerCh, nCh, 2), 256, 0, stream>>>(
            dst, ref, histD, histR, vec4PerThread);
    }
    // 2) CDFs (WMMA cumsum) + transfer tables
    hm_tables_kernel<<<nCh, 32, 0, stream>>>(histD, histR, tables,
                                             1.0f / (float)HW);
    // 3) apply LUT
    hm_apply_kernel<<<dim3(HW / 2048, nCh), 256, 0, stream>>>(dst, tables, out);
}